// YOLONMSLayer_29557964931607
// MI455X (gfx1250) — compile-verified
//
#include <hip/hip_runtime.h>
#include <hip/hip_bf16.h>
#include <stdint.h>

// ---------------- problem constants (from reference) ----------------
#define N_BOXES     10647
#define NUM_CLASSES 80
#define MAX_BOXES   20
#define IOU_THR     0.1f
#define SCORE_THR   0.3f
#define BLOCK       256   // 8 wave32s per workgroup

// flat output layout: [boxes | scores^T | nms_table]
#define OUT_SCORES_OFF (N_BOXES * 4)                               // 42588
#define OUT_NMS_OFF    (OUT_SCORES_OFF + NUM_CLASSES * N_BOXES)    // 894348

// boxes copy chunk per workgroup (ceil(10647/80) = 134 float4s)
#define COPY_CHUNK ((N_BOXES + NUM_CLASSES - 1) / NUM_CLASSES)

typedef unsigned int u32;
typedef __attribute__((ext_vector_type(4))) u32 v4u;
typedef __attribute__((ext_vector_type(4))) int v4i;
typedef __attribute__((ext_vector_type(8))) int v8i;

__device__ __forceinline__ float neg_inf() { return __int_as_float(0xFF800000); }

// Single fused kernel: one workgroup per class.
//  - copies its 1/80th chunk of the boxes passthrough output
//  - TDM-gathers its strided score column into LDS (async tensor op)
//  - emits the transposed-scores output from the staged LDS column
//  - runs the 20-step greedy NMS chain on the LDS scores
__global__ __launch_bounds__(BLOCK) void yolo_nms_kernel(
    const float4* __restrict__ boxes4,
    const float*  __restrict__ scores,   // [N_BOXES, NUM_CLASSES] row-major
    float* __restrict__ out) {
  __shared__ float sScores[N_BOXES];     // 42,588 B — class-c score column
  __shared__ float sRedV[BLOCK / 32];
  __shared__ int   sRedI[BLOCK / 32];
  __shared__ int   sJ;
  __shared__ float sV;

  const int c    = blockIdx.x;
  const int tid  = threadIdx.x;
  const int wave = tid >> 5;
  const int lane = tid & 31;

#if defined(__HIP_DEVICE_COMPILE__) && __has_builtin(__builtin_amdgcn_tensor_load_to_lds)
  // One wave issues a single TDM descriptor: 2D tile [1 x 10647] elements of
  // 4 bytes, row stride = NUM_CLASSES elements -> contiguous column in LDS.
  if (tid < 32) {
    u32 lds_addr = (u32)(uintptr_t)(&sScores[0]);
    unsigned long long ga = (unsigned long long)(uintptr_t)(scores + c);
    // D# group 0: count=1 | lds_addr | global_addr[56:0] | type=2 ("image")
    v4u g0 = { 1u,
               lds_addr,
               (u32)ga,
               (u32)((ga >> 32) & 0x1FFFFFFull) | (2u << 30) };
    // D# group 1: data_size=2 (4B), tensor_dim0=80, tensor_dim1=10647,
    //             tile_dim0=1, tile_dim1=10647, tensor_dim0_stride=80
    v8i g1 = { (int)(2u << 16),                   // [17:16] data_size
               (int)((u32)NUM_CLASSES << 16),     // [63:48] tensor_dim0 lo
               (int)((u32)N_BOXES << 16),         // [95:80] tensor_dim1 lo
               (int)(1u << 16),                   // [127:112] tile_dim0 = 1
               (int)N_BOXES,                      // [143:128] tile_dim1
               (int)NUM_CLASSES,                  // [191:160] dim0_stride lo
               0, 0 };
    v4i z4 = { 0, 0, 0, 0 };
#if defined(__clang_major__) && (__clang_major__ >= 23)
    v8i z8 = { 0, 0, 0, 0, 0, 0, 0, 0 };
    __builtin_amdgcn_tensor_load_to_lds(g0, g1, z4, z4, z8, 0);
#else
    __builtin_amdgcn_tensor_load_to_lds(g0, g1, z4, z4, 0);
#endif
  }
#endif

  // Overlap with the DMA: copy this block's chunk of the boxes passthrough
  // output, and warm WGP$ with the full boxes array (used 20x below).
  {
    const int start = c * COPY_CHUNK;
    const int end   = (start + COPY_CHUNK < N_BOXES) ? start + COPY_CHUNK
                                                     : N_BOXES;
    float4* outB = (float4*)out;
    for (int i = start + tid; i < end; i += BLOCK) outB[i] = boxes4[i];
    for (int i = tid; i < N_BOXES; i += BLOCK)
      __builtin_prefetch(&boxes4[i], 0, 3);
  }

#if defined(__HIP_DEVICE_COMPILE__) && __has_builtin(__builtin_amdgcn_tensor_load_to_lds)
  if (tid < 32)
    __builtin_amdgcn_s_wait_tensorcnt(0);        // issuing wave's DMA done
#else
  // Fallback (host pass / exotic toolchain): direct strided gather.
  for (int i = tid; i < N_BOXES; i += BLOCK)
    sScores[i] = scores[(size_t)i * NUM_CLASSES + c];
#endif
  __syncthreads();                               // publish LDS to all 8 waves

  // Output #2: transposed scores straight from the staged LDS column,
  // then threshold the LDS copy in place (reference: s0 = v>=thr ? v : -inf).
  float* outT = out + OUT_SCORES_OFF + (size_t)c * N_BOXES;
  for (int i = tid; i < N_BOXES; i += BLOCK) {
    float v = sScores[i];
    outT[i]    = v;
    sScores[i] = (v >= SCORE_THR) ? v : neg_inf();
  }
  __syncthreads();

  float* nmsOut = out + OUT_NMS_OFF + (size_t)c * MAX_BOXES * 3;

  for (int it = 0; it < MAX_BOXES; ++it) {
    // ---- argmax over LDS (first-max tie-break to match jnp.argmax) ----
    float bv = neg_inf();
    int   bi = -1;
    for (int i = tid; i < N_BOXES; i += BLOCK) {
      float v = sScores[i];
      if (v > bv) { bv = v; bi = i; }           // strictly > keeps lowest idx
    }
#pragma unroll
    for (int off = 16; off > 0; off >>= 1) {    // wave32 shuffle reduction
      float v2 = __shfl_down(bv, off, 32);
      int   i2 = __shfl_down(bi, off, 32);
      if (v2 > bv || (v2 == bv && (u32)i2 < (u32)bi)) { bv = v2; bi = i2; }
    }
    if (lane == 0) { sRedV[wave] = bv; sRedI[wave] = bi; }
    __syncthreads();
    if (tid == 0) {
      float v = sRedV[0]; int j = sRedI[0];
#pragma unroll
      for (int w = 1; w < BLOCK / 32; ++w) {
        float v2 = sRedV[w]; int i2 = sRedI[w];
        if (v2 > v || (v2 == v && (u32)i2 < (u32)j)) { v = v2; j = i2; }
      }
      sJ = j; sV = v;
    }
    __syncthreads();

    int   j    = sJ;
    float vmax = sV;
#if defined(__HIP_DEVICE_COMPILE__)
    j = __builtin_amdgcn_readfirstlane(j);      // scalarize winner index
#endif
    bool ok = vmax > neg_inf();

    // ---- suppress: IoU(winner, all) > thr, plus the winner itself ----
    if (ok) {
      float4 wb = boxes4[j];                    // uniform -> scalar load
      float  wA = (wb.z - wb.x) * (wb.w - wb.y);
      for (int i = tid; i < N_BOXES; i += BLOCK) {
        float4 b  = boxes4[i];                  // coalesced 512B/wave, in WGP$
        float t0 = fmaxf(wb.x, b.x);
        float t1 = fmaxf(wb.y, b.y);
        float b0 = fminf(wb.z, b.z);
        float b1 = fminf(wb.w, b.w);
        float inter = fmaxf(b0 - t0, 0.0f) * fmaxf(b1 - t1, 0.0f);
        float a2    = (b.z - b.x) * (b.w - b.y);
        float uni   = wA + a2 - inter;
        float iou   = (uni > 0.0f) ? (inter / uni) : 0.0f;
        if (iou > IOU_THR || i == j) sScores[i] = neg_inf();
      }
    }
    if (tid == 0) {
      nmsOut[it * 3 + 0] = ok ? 0.0f      : -1.0f;
      nmsOut[it * 3 + 1] = ok ? (float)c  : -1.0f;
      nmsOut[it * 3 + 2] = ok ? (float)j  : -1.0f;
    }
    __syncthreads();                            // LDS kills visible next iter
  }
}

extern "C" void kernel_launch(void* const* d_in, const int* in_sizes, int n_in,
                              void* d_out, int out_size, void* d_ws, size_t ws_size,
                              hipStream_t stream) {
  (void)in_sizes; (void)n_in; (void)out_size; (void)d_ws; (void)ws_size;
  const float4* boxes  = (const float4*)d_in[0];
  const float*  scores = (const float*)d_in[1];
  float*        out    = (float*)d_out;

  yolo_nms_kernel<<<NUM_CLASSES, BLOCK, 0, stream>>>(boxes, scores, out);
}